// AttentionModule_68212670595907
// MI455X (gfx1250) — compile-verified
//
#include <hip/hip_runtime.h>
#include <math.h>

typedef float v2f __attribute__((ext_vector_type(2)));
typedef float v8f __attribute__((ext_vector_type(8)));

#define Bv 32
#define Tv 2048
#define Hv 1024
#define CH 16                 // T-chunks per batch
#define TPC (Tv / CH)         // 128 t per block
#define TPW (TPC / 8)         // 16 t per wave (8 waves/block)
#define PSTRIDE (Hv + 2)      // partial: c[1024], m, s

// ---------------- Kernel 1: v[b,h] = sum_o hidden[b,o] * W[o,h] via WMMA f32 16x16x4
// M=32 (2 tiles), N=1024 (64 tiles), K=1024 (256 steps of 4). One 16x16 tile per wave.
__global__ __launch_bounds__(128) void k_gemm_v(const float* __restrict__ hidden,
                                                const float* __restrict__ W,
                                                float* __restrict__ v) {
  const int lane = threadIdx.x & 31;
  const int gw   = blockIdx.x * (blockDim.x >> 5) + (threadIdx.x >> 5); // 0..127
  const int tm   = gw >> 6;        // 0..1   (batch tile)
  const int tn   = gw & 63;        // 0..63  (h tile)
  const int half = lane >> 4;      // K-pair select per ISA 16x4 f32 layout
  const int l15  = lane & 15;
  const int mrow = tm * 16 + l15;  // A: M = lane&15
  const int ncol = tn * 16 + l15;  // B: N = lane&15
  const int kgrp = half * 2;       // lanes 0-15: K+0/K+1 ; lanes 16-31: K+2/K+3

  v8f acc = {};
  for (int k0 = 0; k0 < Hv; k0 += 4) {
    v2f a, b;
    a.x = hidden[(size_t)mrow * Hv + k0 + kgrp];
    a.y = hidden[(size_t)mrow * Hv + k0 + kgrp + 1];
    b.x = W[(size_t)(k0 + kgrp) * Hv + ncol];
    b.y = W[(size_t)(k0 + kgrp + 1) * Hv + ncol];
    acc = __builtin_amdgcn_wmma_f32_16x16x4_f32(false, a, false, b,
                                                (short)0, acc, false, false);
  }
  // D layout: VGPR r -> M = r (+8 for lanes 16-31), N = lane&15
  const int mbase = tm * 16 + half * 8;
#pragma unroll
  for (int r = 0; r < 8; ++r)
    v[(size_t)(mbase + r) * Hv + ncol] = acc[r];
}

// ---------------- Kernel 2: fused score + online softmax + context partials
// grid (CH, B), 256 threads. enc read from HBM exactly once.
__global__ __launch_bounds__(256) void k_attn(const float* __restrict__ enc,
                                              const float* __restrict__ v,
                                              float* __restrict__ scores,
                                              float* __restrict__ part) {
  const int b     = blockIdx.y;
  const int chunk = blockIdx.x;
  const int tid   = threadIdx.x;
  const int wave  = tid >> 5;
  const int lane  = tid & 31;

  __shared__ float v_s[Hv];
  __shared__ float c_s[Hv];
  __shared__ float m_s[8], s_s[8];

  for (int i = tid; i < Hv; i += 256) { v_s[i] = v[(size_t)b * Hv + i]; c_s[i] = 0.f; }
  __syncthreads();

  float m = -INFINITY, s = 0.f;
  float4 c[8];
#pragma unroll
  for (int j = 0; j < 8; ++j) c[j] = make_float4(0.f, 0.f, 0.f, 0.f);

  const int t0 = chunk * TPC + wave * TPW;
  const float4* v4 = (const float4*)v_s;

  for (int tt = 0; tt < TPW; ++tt) {
    const int t = t0 + tt;
    const float* __restrict__ row = enc + ((size_t)b * Tv + t) * Hv;
    if (tt + 1 < TPW)
      __builtin_prefetch(row + Hv + lane * 32, 0, 0); // global_prefetch_b8, next row

    const float4* row4 = (const float4*)row;
    float4 x[8];
    float p = 0.f;
#pragma unroll
    for (int j = 0; j < 8; ++j) {               // 8 x b128 loads per lane, coalesced
      x[j] = row4[lane + 32 * j];
      float4 vv = v4[lane + 32 * j];
      p += x[j].x * vv.x + x[j].y * vv.y + x[j].z * vv.z + x[j].w * vv.w;
    }
#pragma unroll
    for (int off = 16; off > 0; off >>= 1) p += __shfl_xor(p, off, 32);
    if (lane == 0) scores[(size_t)b * Tv + t] = p;

    // online softmax; p is wave-uniform -> branch is uniform
    float w;
    if (p > m) {
      const float scale = __expf(m - p);        // first iter: exp(-inf)=0
      s = s * scale + 1.f;
#pragma unroll
      for (int j = 0; j < 8; ++j) {
        c[j].x *= scale; c[j].y *= scale; c[j].z *= scale; c[j].w *= scale;
      }
      m = p; w = 1.f;
    } else {
      w = __expf(p - m);
      s += w;
    }
#pragma unroll
    for (int j = 0; j < 8; ++j) {               // reuse loaded values: no second read
      c[j].x += w * x[j].x; c[j].y += w * x[j].y;
      c[j].z += w * x[j].z; c[j].w += w * x[j].w;
    }
  }

  if (lane == 0) { m_s[wave] = m; s_s[wave] = s; }
  __syncthreads();

  float Mb = -INFINITY;
#pragma unroll
  for (int wv = 0; wv < 8; ++wv) Mb = fmaxf(Mb, m_s[wv]);
  float Sb = 0.f;
#pragma unroll
  for (int wv = 0; wv < 8; ++wv) Sb += s_s[wv] * __expf(m_s[wv] - Mb);

  const float scw = __expf(m - Mb);             // rescale this wave's contribution
#pragma unroll
  for (int j = 0; j < 8; ++j) {
    const int h = 4 * (lane + 32 * j);
    atomicAdd(&c_s[h + 0], scw * c[j].x);       // ds_add_f32
    atomicAdd(&c_s[h + 1], scw * c[j].y);
    atomicAdd(&c_s[h + 2], scw * c[j].z);
    atomicAdd(&c_s[h + 3], scw * c[j].w);
  }
  __syncthreads();

  float* pc = part + (size_t)(b * CH + chunk) * PSTRIDE;
  for (int i = tid; i < Hv; i += 256) pc[i] = c_s[i];
  if (tid == 0) { pc[Hv] = Mb; pc[Hv + 1] = Sb; }
}

// ---------------- Kernel 3: merge chunk partials; write context + normalized weights
__global__ __launch_bounds__(256) void k_final(const float* __restrict__ part,
                                               const float* __restrict__ scores,
                                               float* __restrict__ out) {
  const int b = blockIdx.x;
  const int tid = threadIdx.x;

  float M = -INFINITY;
#pragma unroll
  for (int k = 0; k < CH; ++k)
    M = fmaxf(M, part[(size_t)(b * CH + k) * PSTRIDE + Hv]);

  float ek[CH];
  float S = 0.f;
#pragma unroll
  for (int k = 0; k < CH; ++k) {
    const float* pk = part + (size_t)(b * CH + k) * PSTRIDE;
    ek[k] = __expf(pk[Hv] - M);
    S += pk[Hv + 1] * ek[k];
  }
  const float invS = 1.f / S;

  float* ctx  = out;                 // [B, H]
  float* attn = out + Bv * Hv;       // [B, T]

  for (int h = tid; h < Hv; h += 256) {
    float acc = 0.f;
#pragma unroll
    for (int k = 0; k < CH; ++k)
      acc += ek[k] * part[(size_t)(b * CH + k) * PSTRIDE + h];
    ctx[(size_t)b * Hv + h] = acc * invS;
  }
  for (int t = tid; t < Tv; t += 256)
    attn[(size_t)b * Tv + t] = __expf(scores[(size_t)b * Tv + t] - M) * invS;
}

extern "C" void kernel_launch(void* const* d_in, const int* in_sizes, int n_in,
                              void* d_out, int out_size, void* d_ws, size_t ws_size,
                              hipStream_t stream) {
  const float* hidden = (const float*)d_in[0]; // [32,1024]
  const float* enc    = (const float*)d_in[1]; // [32,2048,1024]
  const float* W      = (const float*)d_in[2]; // [1024,1024]
  float* out = (float*)d_out;

  float* ws     = (float*)d_ws;
  float* v      = ws;                 // 32*1024
  float* scores = v + Bv * Hv;        // 32*2048
  float* part   = scores + Bv * Tv;   // 32*16*1026

  k_gemm_v<<<32, 128, 0, stream>>>(hidden, W, v);
  dim3 g2(CH, Bv);
  k_attn<<<g2, 256, 0, stream>>>(enc, v, scores, part);
  k_final<<<Bv, 256, 0, stream>>>(part, scores, out);
}